// HybridQuanvolution_65481071409993
// MI455X (gfx1250) — compile-verified
//
#include <hip/hip_runtime.h>

typedef __attribute__((ext_vector_type(2))) float v2f;
typedef __attribute__((ext_vector_type(8))) float v8f;

#define WAVES_PER_BLOCK 8
#define IMG_PER_WAVE    16
#define IMG_PER_BLOCK   (WAVES_PER_BLOCK * IMG_PER_WAVE)   // 128

__global__ __launch_bounds__(256)
void quanv_fused_wmma(const float* __restrict__ x,
                      const float* __restrict__ params,
                      const float* __restrict__ W,
                      const float* __restrict__ b,
                      float* __restrict__ out,
                      int nImg)
{
    // W^T padded to 16 classes, [k][n] layout: 784*16 f32 = 50176 B
    __shared__ float ldsW[784 * 16];
    // per-wave 16x16 logits tile, padded to 17 to dodge bank conflicts
    __shared__ float sm[WAVES_PER_BLOCK][16][17];

    const int tid  = threadIdx.x;
    const int wave = tid >> 5;
    const int lane = tid & 31;
    const int half = lane >> 4;    // 0: K=0,1 / 1: K=2,3 within each patch
    const int nsel = lane & 15;    // M index for A, N index for B/C/D

    // ---- stage W^T (zero-padded N=10..15) into LDS ----
    for (int idx = tid; idx < 784 * 16; idx += 256) {
        int k = idx >> 4, n = idx & 15;
        ldsW[idx] = (n < 10) ? W[n * 784 + k] : 0.0f;
    }
    __syncthreads();

    const int tileBase = blockIdx.x * IMG_PER_BLOCK + wave * IMG_PER_WAVE;
    int img = tileBase + nsel;
    if (img >= nImg) img = nImg - 1;          // clamp loads (B % 128 == 0 in practice)

    const float p0 = params[0], p1 = params[1], p2 = params[2], p3 = params[3];
    const float pA = half ? p2 : p0;
    const float pB = half ? p3 : p1;

    // this lane reads pixel pairs from row (2*pr + half) of its image
    const float* xr = x + (size_t)img * 784 + half * 28;

    v8f acc = {};

    int p = 0;
    for (int pr = 0; pr < 14; ++pr) {
        const float* row = xr + pr * 56;      // 2 image rows per patch-row
        for (int pc = 0; pc < 14; ++pc, ++p) {
            v2f e = *(const v2f*)(row + pc * 2);
            // RY closed form: z = cos(pixel + param)
            float zf = __cosf(e.x + pA);
            float zs = __cosf(e.y + pB);
            v2f a;
            a.x = zf;          // K = 2*half + 0  ->  z0  (or z2)
            a.y = zf * zs;     // K = 2*half + 1  ->  z0*z1 (or z2*z3)

            // B fragment: Wt[4p + 2*half + {0,1}][nsel]
            int kb = (p * 4 + 2 * half) * 16 + nsel;
            v2f bf;
            bf.x = ldsW[kb];
            bf.y = ldsW[kb + 16];

            acc = __builtin_amdgcn_wmma_f32_16x16x4_f32(
                    /*neg_a=*/false, a, /*neg_b=*/false, bf,
                    /*c_mod=*/(short)0, acc,
                    /*reuse_a=*/false, /*reuse_b=*/false);
        }
    }

    // ---- bias (zero for padded classes) ----
    float bv = (nsel < 10) ? b[nsel] : 0.0f;
    #pragma unroll
    for (int i = 0; i < 8; ++i) acc[i] += bv;

    // ---- spill D tile: row M = i + 8*half, col N = nsel ----
    #pragma unroll
    for (int i = 0; i < 8; ++i)
        sm[wave][i + 8 * half][nsel] = acc[i];

    __syncthreads();   // make cross-lane LDS writes visible (also orders vs compiler)

    // ---- log-softmax: lanes 0..15 each handle one image's 10 logits ----
    if (lane < 16) {
        int m    = lane;
        int oimg = tileBase + m;
        if (oimg < nImg) {
            float t[10];
            float mx = -3.402823466e38f;
            #pragma unroll
            for (int n = 0; n < 10; ++n) {
                t[n] = sm[wave][m][n];
                mx = fmaxf(mx, t[n]);
            }
            float s = 0.0f;
            #pragma unroll
            for (int n = 0; n < 10; ++n) s += __expf(t[n] - mx);
            float lse = mx + __logf(s);
            float* o = out + (size_t)oimg * 10;
            #pragma unroll
            for (int n = 0; n < 10; ++n) o[n] = t[n] - lse;
        }
    }
}

extern "C" void kernel_launch(void* const* d_in, const int* in_sizes, int n_in,
                              void* d_out, int out_size, void* d_ws, size_t ws_size,
                              hipStream_t stream)
{
    const float* x      = (const float*)d_in[0];   // [B, 28, 28]
    const float* params = (const float*)d_in[1];   // [4]
    const float* W      = (const float*)d_in[2];   // [10, 784]
    const float* b      = (const float*)d_in[3];   // [10]
    float* out          = (float*)d_out;           // [B, 10]

    int nImg = in_sizes[0] / 784;
    int blocks = (nImg + IMG_PER_BLOCK - 1) / IMG_PER_BLOCK;

    quanv_fused_wmma<<<blocks, 256, 0, stream>>>(x, params, W, b, out, nImg);
}